// MaskModule_11759620456534
// MI455X (gfx1250) — compile-verified
//
#include <hip/hip_runtime.h>

typedef __attribute__((ext_vector_type(2))) float v2f;
typedef __attribute__((ext_vector_type(8))) float v8f;

#if __has_builtin(__builtin_amdgcn_global_load_async_to_lds_b32)
#define HAVE_ASYNC_LDS 1
#endif

typedef __attribute__((address_space(1))) int* gas_int;   // global (AS1) src
typedef __attribute__((address_space(3))) int* las_int;   // LDS (AS3) dst

// ---------------- filters (fp32, from reference) ----------------
__constant__ float c_h[9] = {
    0.037828455506995f, -0.02384946501938f, -0.11062440441842f, 0.37740285561265f,
    0.8526986790094f,
    0.37740285561265f, -0.11062440441842f, -0.02384946501938f, 0.037828455506995f};
__constant__ float c_g[7] = {
    -0.064538882628938f, -0.040689417609558f, 0.41809227322221f, 0.78848561640566f,
    0.41809227322221f, -0.040689417609558f, -0.064538882628938f};
__constant__ float c_f[12] = {
    0.0144f, 0.0272f, 0.0526f, 0.0972f, 0.193f, 0.63f,
    -0.63f, -0.193f, -0.0972f, -0.0526f, -0.0272f, -0.0144f};

// ---------------- stage A: c = (h x h * x)[::2,::2] ----------------
__global__ void k_h_rows_dec(const float* __restrict__ x, float* __restrict__ t1) {
    int idx = blockIdx.x * blockDim.x + threadIdx.x;   // 256*256*128
    int p = idx >> 15; int rem = idx & 32767; int r = rem >> 7; int j = rem & 127;
    const float* xp = x + (size_t)p * 65536 + (size_t)r * 256;
    int base = 2 * j - 4;
    float s = 0.f;
#pragma unroll
    for (int v = 0; v < 9; ++v) s += c_h[v] * xp[(base + v) & 255];
    t1[(size_t)p * 32768 + r * 128 + j] = s;
}

__global__ void k_h_cols_dec(const float* __restrict__ t1, float* __restrict__ cws,
                             float* __restrict__ outc) {
    int idx = blockIdx.x * blockDim.x + threadIdx.x;   // 256*128*128
    int p = idx >> 14; int i = (idx >> 7) & 127; int j = idx & 127;
    const float* tp = t1 + (size_t)p * 32768;
    int base = 2 * i - 4;
    float s = 0.f;
#pragma unroll
    for (int u = 0; u < 9; ++u) s += c_h[u] * tp[((base + u) & 255) * 128 + j];
    size_t o = (size_t)p * 16384 + i * 128 + j;
    cws[o] = s;
    outc[o] = s;
}

// ---------------- stage B: d = x - g x g * upsample2(c) ----------------
__global__ void k_g_rows_up(const float* __restrict__ cws, float* __restrict__ rc) {
    int idx = blockIdx.x * blockDim.x + threadIdx.x;   // 256*128*256
    int p = idx >> 15; int i = (idx >> 8) & 127; int cc = idx & 255;
    const float* cp = cws + (size_t)p * 16384 + (size_t)i * 128;
    float s = 0.f;
#pragma unroll
    for (int v = 0; v < 7; ++v) {
        int col = (cc - 3 + v) & 255;
        if ((col & 1) == 0) s += c_g[v] * cp[col >> 1];
    }
    rc[(size_t)p * 32768 + i * 256 + cc] = s;
}

__global__ void k_g_cols_d(const float* __restrict__ x, const float* __restrict__ rc,
                           float* __restrict__ d) {
    int idx = blockIdx.x * blockDim.x + threadIdx.x;   // 256*256*256
    int p = idx >> 16; int r = (idx >> 8) & 255; int cc = idx & 255;
    const float* rcp = rc + (size_t)p * 32768;
    float s = 0.f;
#pragma unroll
    for (int u = 0; u < 7; ++u) {
        int rr = (r - 3 + u) & 255;
        if ((rr & 1) == 0) s += c_g[u] * rcp[(rr >> 1) * 256 + cc];
    }
    size_t o = (size_t)p * 65536 + (size_t)r * 256 + cc;
    d[o] = x[o] - s;
}

// ---------------- shear gathers ----------------
__global__ void k_shear1(const float* __restrict__ d, float* __restrict__ p0,
                         float* __restrict__ p1) {
    int idx = blockIdx.x * blockDim.x + threadIdx.x;   // 256*128*256
    int p = idx >> 15; int a = (idx >> 8) & 127; int b = idx & 255;
    int row = (a - b) & 255, c1 = (a + b) & 255, c2 = (a + b + 1) & 255;
    const float* dp = d + (size_t)p * 65536 + (size_t)row * 256;
    size_t o = (size_t)p * 32768 + a * 256 + b;
    p0[o] = dp[c1];
    p1[o] = dp[c2];
}

__global__ void k_shear2(const float* __restrict__ y0, const float* __restrict__ y1,
                         float* __restrict__ p0, float* __restrict__ p1) {
    int idx = blockIdx.x * blockDim.x + threadIdx.x;   // 512*128*128
    int q = idx >> 14; int i = (idx >> 7) & 127; int j = idx & 127;
    int n = q >> 7, ch = q & 127;
    const float* zp = (ch < 64) ? (y0 + ((size_t)(n * 64 + ch)) * 32768)
                                : (y1 + ((size_t)(n * 64 + (ch - 64))) * 32768);
    int r1 = (i + j) & 127, r2 = (i + j + 1) & 127, cc = (j - i) & 255;
    size_t o = (size_t)q * 16384 + i * 128 + j;
    p0[o] = zp[r1 * 256 + cc];
    p1[o] = zp[r2 * 256 + cc];
}

// ---------------- WMMA separable 12-tap conv + combine ----------------
// out = alpha*aux + beta*(f (x) f * in), periodic / qper_col wrap.
// 32x32 output tile per 128-thread (4-wave) block.
//   row pass : D = Afilt(16x28) x Bdata(28x16), Afilt[m,k]=f[k-m]
//   col pass : D = Adata(16x28) x Bfilt(28x16), Bfilt[k,n]=f[k-n]
__global__ __launch_bounds__(128) void k_sepconv_f(
    const float* __restrict__ in, const float* __restrict__ aux,
    float* __restrict__ outc, float* __restrict__ out_lo, float* __restrict__ out_hi,
    int H, int W, int qper, int roff, int coff, float alpha, float beta) {
    __shared__ __align__(16) float ext[44 * 48];  // haloed tile (43x43 valid, 0-padded)
    __shared__ __align__(16) float rp[32 * 48];   // row-pass result
    __shared__ __align__(16) float fpad[48];      // zero-padded filter: fpad[i]=f[i-15]
    const int EC = 48, RC = 48;

    int plane = blockIdx.z;
    int rb = blockIdx.y * 32, cb = blockIdx.x * 32;
    size_t pstr = (size_t)H * W;
    const float* inp = in + (size_t)plane * pstr;
    int tid = threadIdx.x;

    if (tid < 48) {
        int d0 = tid - 15;
        fpad[tid] = ((unsigned)d0 < 12u) ? c_f[d0] : 0.0f;
    }

    // ---- stage haloed tile into LDS (async DMA gather if available) ----
    for (int i = tid; i < 44 * 48; i += 128) {
        int er = i / EC, ec = i - er * EC;
        if (er < 43 && ec < 43) {
            int gr = rb - roff + er;
            int gc = cb - coff + ec;
            int sr = gr, sh = 0;
            if (sr < 0)       { sr += H; sh = qper ? (W >> 1) : 0; }
            else if (sr >= H) { sr -= H; sh = qper ? (W >> 1) : 0; }
            int sc = (gc + sh + W) & (W - 1);
            const float* src = inp + (size_t)sr * W + sc;
#ifdef HAVE_ASYNC_LDS
            __builtin_amdgcn_global_load_async_to_lds_b32(
                (gas_int)(int*)src, (las_int)(int*)&ext[i], 0, 0);
#else
            ext[i] = *src;
#endif
        } else {
            ext[i] = 0.f;   // pad: disjoint from async targets, plain DS store
        }
    }
#ifdef HAVE_ASYNC_LDS
#if __has_builtin(__builtin_amdgcn_s_wait_asynccnt)
    __builtin_amdgcn_s_wait_asynccnt(0);
#else
    asm volatile("s_wait_asynccnt 0" ::: "memory");
#endif
#endif
    __syncthreads();

    int wave = tid >> 5, lane = tid & 31;
    int half = lane >> 4, sub = lane & 15;

    // per-lane banded-filter operand values, kept in registers for both passes
    v2f fw[7];
#pragma unroll
    for (int kc = 0; kc < 7; ++kc) {
        int kA = kc * 4 + half * 2;
        fw[kc].x = fpad[kA - sub + 15];
        fw[kc].y = fpad[kA + 1 - sub + 15];
    }

    // ---- row pass: 2 row-chunks x 3 col-chunks, round-robin over waves ----
#pragma unroll
    for (int job = wave; job < 6; job += 4) {
        int mb = (job / 3) * 16, cb16 = (job % 3) * 16;
        v8f acc = {0.f, 0.f, 0.f, 0.f, 0.f, 0.f, 0.f, 0.f};
#pragma unroll
        for (int kc = 0; kc < 7; ++kc) {
            int kA = kc * 4 + half * 2;
            v2f b;
            b.x = ext[(mb + kA) * EC + cb16 + sub];       // Bdata[kA,   N=sub]
            b.y = ext[(mb + kA + 1) * EC + cb16 + sub];   // Bdata[kA+1, N=sub]
            acc = __builtin_amdgcn_wmma_f32_16x16x4_f32(
                false, fw[kc], false, b, (short)0, acc, false, false);
        }
#pragma unroll
        for (int i2 = 0; i2 < 8; ++i2)
            rp[(mb + half * 8 + i2) * RC + cb16 + sub] = acc[i2];
    }
    __syncthreads();

    // ---- col pass: each wave owns one 16x16 output subtile ----
    {
        int mb = (wave >> 1) * 16, nb = (wave & 1) * 16;
        v8f acc = {0.f, 0.f, 0.f, 0.f, 0.f, 0.f, 0.f, 0.f};
#pragma unroll
        for (int kc = 0; kc < 7; ++kc) {
            int kA = kc * 4 + half * 2;                    // even -> 8B aligned
            v2f a = *(const v2f*)&rp[(mb + sub) * RC + nb + kA];  // Adata[M=sub, kA..kA+1]
            acc = __builtin_amdgcn_wmma_f32_16x16x4_f32(
                false, a, false, fw[kc], (short)0, acc, false, false);
        }
        const float* auxp = aux + (size_t)plane * pstr;
        int gr0 = rb + mb + half * 8;
        int gc = cb + nb + sub;
        float* dstc = outc ? outc + (size_t)plane * pstr : nullptr;
        float* dstp = nullptr;
        if (out_lo) {   // channel-split permuted write for e0/e1 (512-plane stage)
            int n = plane >> 7, ch = plane & 127;
            dstp = (ch < 64) ? out_lo + (size_t)(n * 64 + ch) * pstr
                             : out_hi + (size_t)(n * 64 + (ch - 64)) * pstr;
        }
#pragma unroll
        for (int i2 = 0; i2 < 8; ++i2) {
            int gr = gr0 + i2;
            float vv = alpha * auxp[(size_t)gr * W + gc] + beta * acc[i2];
            if (dstc) dstc[(size_t)gr * W + gc] = vv;
            if (dstp) dstp[(size_t)gr * W + gc] = vv;
        }
    }
}

// ---------------- host side ----------------
extern "C" void kernel_launch(void* const* d_in, const int* in_sizes, int n_in,
                              void* d_out, int out_size, void* d_ws, size_t ws_size,
                              hipStream_t stream) {
    (void)in_sizes; (void)n_in; (void)out_size; (void)ws_size;
    const float* x = (const float*)d_in[0];          // (4,64,256,256)
    float* out = (float*)d_out;

    // workspace slots (floats)
    float* S0 = (float*)d_ws;            // 16,777,216 : d  -> e0
    float* S1 = S0 + 16777216;           //  8,388,608 : t1 -> p0 -> p0'
    float* S2 = S1 + 8388608;            //  8,388,608 : rc -> p1 -> p1'
    float* S3 = S2 + 8388608;            //  4,194,304 : c
    float* S4 = S3 + 4194304;            //  8,388,608 : y0
    float* S5 = S4 + 8388608;            //  8,388,608 : y1

    const int CH = 4194304;              // 4*64*128*128
    float* out_c    = out;
    float* out_e1lo = out + (size_t)CH;
    float* out_e0lo = out + (size_t)2 * CH;
    float* out_e0hi = out + (size_t)3 * CH;
    float* out_e1hi = out + (size_t)4 * CH;

    const float INV_SQ2 = 0.70710678118654752f;
    const float SQ2     = 1.41421356237309515f;

    // stage A: c
    k_h_rows_dec<<<8388608 / 256, 256, 0, stream>>>(x, S1);
    k_h_cols_dec<<<4194304 / 256, 256, 0, stream>>>(S1, S3, out_c);
    // stage B: d
    k_g_rows_up<<<8388608 / 256, 256, 0, stream>>>(S3, S2);
    k_g_cols_d<<<16777216 / 256, 256, 0, stream>>>(x, S2, S0);
    // stage C: quincunx shear then fbdec with qper_col
    k_shear1<<<8388608 / 256, 256, 0, stream>>>(S0, S1, S2);
    k_sepconv_f<<<dim3(8, 4, 256), 128, 0, stream>>>(
        S2 /*in=p1*/, S1 /*aux=p0*/, S4 /*y0*/, nullptr, nullptr,
        128, 256, 1, 6, 6, INV_SQ2, -INV_SQ2);
    k_sepconv_f<<<dim3(8, 4, 256), 128, 0, stream>>>(
        S4 /*in=y0*/, S2 /*aux=p1*/, S5 /*y1*/, nullptr, nullptr,
        128, 256, 1, 5, 5, -SQ2, -1.0f);
    // stage D: shear of z=concat(y0,y1), fbdec with per extension, split writes
    k_shear2<<<8388608 / 256, 256, 0, stream>>>(S4, S5, S1, S2);
    k_sepconv_f<<<dim3(4, 4, 512), 128, 0, stream>>>(
        S2 /*in=p1'*/, S1 /*aux=p0'*/, S0 /*e0*/, out_e0lo, out_e0hi,
        128, 128, 0, 6, 6, INV_SQ2, -INV_SQ2);
    k_sepconv_f<<<dim3(4, 4, 512), 128, 0, stream>>>(
        S0 /*in=e0*/, S2 /*aux=p1'*/, nullptr, out_e1lo, out_e1hi,
        128, 128, 0, 5, 5, -SQ2, -1.0f);
}